// UltraPrecisionModule_3659312136248
// MI455X (gfx1250) — compile-verified
//
#include <hip/hip_runtime.h>
#include <stdint.h>

#ifndef __has_builtin
#define __has_builtin(x) 0
#endif

#define AS1 __attribute__((address_space(1)))
#define AS3 __attribute__((address_space(3)))

#define BLOCK    64
#define BINS     256
#define WIMG     256
#define HW       (256 * 256)
#define NBLOCKS  8192   // 8*64*4*4 tiles
#define NCH      512    // 8*64 channels
#define TILE_N   4096   // 64*64 elements per tile

typedef __attribute__((ext_vector_type(2))) float v2f;
typedef __attribute__((ext_vector_type(8))) float v8f;
typedef __attribute__((ext_vector_type(4))) int   v4i;

// ---- CDNA5 async global->LDS copy (ASYNCcnt-tracked), 16B per lane ----
__device__ __forceinline__ void async_ld_b128(const float* g, float* l) {
#if __has_builtin(__builtin_amdgcn_global_load_async_to_lds_b128)
  __builtin_amdgcn_global_load_async_to_lds_b128(
      (AS1 v4i*)(void*)g, (AS3 v4i*)(void*)l, 0, 0);
#else
  asm volatile("global_load_async_to_lds_b128 %0, %1, off"
               :: "v"((AS3 float*)l), "v"(g)
               : "memory");
#endif
}

__device__ __forceinline__ void wait_async0() {
#if __has_builtin(__builtin_amdgcn_s_wait_asynccnt)
  __builtin_amdgcn_s_wait_asynccnt(0);
#else
  asm volatile("s_wait_asynccnt 0x0" ::: "memory");
#endif
}

// =====================================================================
// Kernel 1: per-64x64-tile Shannon entropy (256-bin histogram)
// One workgroup per tile; 256 threads; everything staged in LDS.
// =====================================================================
__global__ __launch_bounds__(256) void entropy_kernel(const float* __restrict__ x,
                                                      float* __restrict__ ent) {
  __shared__ float    sdata[TILE_N];   // 16 KB tile stage
  __shared__ float    red[256];        // reduction scratch
  __shared__ unsigned hist[BINS];      // histogram bins
  __shared__ float    s_bmin, s_bmax;

  const int b   = blockIdx.x;
  const int tid = threadIdx.x;
  const int nc  = b >> 4;          // channel index (n*64+c)
  const int t16 = b & 15;
  const int th  = t16 >> 2;        // tile row 0..3
  const int tw  = t16 & 3;         // tile col 0..3
  const float* gbase = x + (size_t)nc * HW + th * (BLOCK * WIMG) + tw * BLOCK;

  // Async-copy 4096 floats (1024 x b128) into LDS.
#pragma unroll
  for (int i = 0; i < 4; ++i) {
    const int e   = i * 256 + tid;   // float4 slot 0..1023
    const int row = e >> 4;          // 16 float4 per 64-wide tile row
    const int c4  = e & 15;
    async_ld_b128(gbase + row * WIMG + c4 * 4, &sdata[e * 4]);
  }
  wait_async0();
  __syncthreads();

  // Each thread owns 16 values (stride-256 -> conflict-free LDS reads).
  float v[16];
  float lmin =  3.402823466e38f;
  float lmax = -3.402823466e38f;
#pragma unroll
  for (int j = 0; j < 16; ++j) {
    v[j] = sdata[tid + j * 256];
    lmin = fminf(lmin, v[j]);
    lmax = fmaxf(lmax, v[j]);
  }

  // min reduction
  red[tid] = lmin; __syncthreads();
  for (int s = 128; s > 0; s >>= 1) {
    if (tid < s) red[tid] = fminf(red[tid], red[tid + s]);
    __syncthreads();
  }
  if (tid == 0) s_bmin = red[0];
  __syncthreads();
  // max reduction
  red[tid] = lmax; __syncthreads();
  for (int s = 128; s > 0; s >>= 1) {
    if (tid < s) red[tid] = fmaxf(red[tid], red[tid + s]);
    __syncthreads();
  }
  if (tid == 0) s_bmax = red[0];
  hist[tid] = 0u;
  __syncthreads();

  const float bmin  = s_bmin;
  const float width = (s_bmax - bmin) * (1.0f / (float)BINS);
  const float wsafe = (width > 0.0f) ? width : 1.0f;

  // Histogram via LDS atomics (matches np.histogram floor+clip semantics).
#pragma unroll
  for (int j = 0; j < 16; ++j) {
    int idx = (int)floorf((v[j] - bmin) / wsafe);
    idx = idx < 0 ? 0 : (idx > BINS - 1 ? BINS - 1 : idx);
    atomicAdd(&hist[idx], 1u);
  }
  __syncthreads();

  // density=True normalization, +EPS, renormalize, entropy.
  const float inv_nw = 1.0f / (4096.0f * wsafe);
  const float p_un   = (float)hist[tid] * inv_nw + 1e-12f;

  red[tid] = p_un; __syncthreads();
  for (int s = 128; s > 0; s >>= 1) {
    if (tid < s) red[tid] += red[tid + s];
    __syncthreads();
  }
  const float S = red[0];
  __syncthreads();

  const float p    = p_un / S;
  const float term = -p * log2f(p);
  red[tid] = term; __syncthreads();
  for (int s = 128; s > 0; s >>= 1) {
    if (tid < s) red[tid] += red[tid + s];
    __syncthreads();
  }
  if (tid == 0) ent[b] = red[0];
}

// =====================================================================
// Kernel 2: bilinear upsample 4x4 -> 256x256 as Out = Wr * E * Wc^T,
// one V_WMMA_F32_16X16X4_F32 per 16x16 output tile.
// =====================================================================

// Half-pixel (align_corners=False) interpolation weights for output index r
// against 4 source samples, edge-clamped. At most 2 nonzero entries.
__device__ __forceinline__ void interp_weights(int r, float w[4]) {
  const float s  = ((float)r + 0.5f) * (1.0f / 64.0f) - 0.5f;  // 4/256
  const float f  = floorf(s);
  const float w1 = s - f;
  const int   i0 = (int)f;
  const int   c0 = i0 < 0 ? 0 : (i0 > 3 ? 3 : i0);
  const int   i1 = i0 + 1;
  const int   c1 = i1 < 0 ? 0 : (i1 > 3 ? 3 : i1);
  w[0] = w[1] = w[2] = w[3] = 0.0f;
  w[c0] += 1.0f - w1;
  w[c1] += w1;
}

__global__ __launch_bounds__(256) void upsample_kernel(const float* __restrict__ ent,
                                                       float* __restrict__ out) {
  __shared__ float sE[16];   // 4x4 entropy map, row-major
  const int ch  = blockIdx.x;
  const int tid = threadIdx.x;
  if (tid < 16) sE[tid] = ent[ch * 16 + tid];
  __syncthreads();

  const int lane  = tid & 31;
  const int wave  = tid >> 5;       // 8 wave32's per workgroup
  const int half  = lane >> 4;      // 0: lanes 0-15, 1: lanes 16-31
  const int l16   = lane & 15;
  const int khalf = half * 2;       // A/B K-pair held by this half-wave

  float* obase = out + (size_t)ch * HW;

  for (int t = wave; t < 256; t += 8) {       // 256 tiles of 16x16
    const int R0 = (t >> 4) << 4;
    const int C0 = (t & 15) << 4;

    // A (16x4 f32): lane layout per ISA — lanes 0-15 hold K=0,1; 16-31 hold K=2,3.
    float wr[4];
    interp_weights(R0 + l16, wr);
    v2f a;
    a.x = wr[khalf];
    a.y = wr[khalf + 1];

    // B (4x16 f32): B[k][n] = sum_j E[k][j] * wc_n[j]  (E * Wc^T column block)
    float wc[4];
    interp_weights(C0 + l16, wc);
    v2f bb;
    bb.x = sE[khalf * 4 + 0] * wc[0] + sE[khalf * 4 + 1] * wc[1] +
           sE[khalf * 4 + 2] * wc[2] + sE[khalf * 4 + 3] * wc[3];
    bb.y = sE[(khalf + 1) * 4 + 0] * wc[0] + sE[(khalf + 1) * 4 + 1] * wc[1] +
           sE[(khalf + 1) * 4 + 2] * wc[2] + sE[(khalf + 1) * 4 + 3] * wc[3];

    v8f c = {0.f, 0.f, 0.f, 0.f, 0.f, 0.f, 0.f, 0.f};
    // D(16x16) = A(16x4) x B(4x16) + 0
    v8f d = __builtin_amdgcn_wmma_f32_16x16x4_f32(
        /*neg_a=*/false, a, /*neg_b=*/false, bb,
        /*c_mod=*/(short)0, c, /*reuse_a=*/false, /*reuse_b=*/false);

    // C/D layout: VGPR v, lane l -> M = v + 8*(l>=16), N = l&15
    const int mbase = half * 8;
#pragma unroll
    for (int vv = 0; vv < 8; ++vv) {
      obase[(size_t)(R0 + mbase + vv) * WIMG + (C0 + l16)] = d[vv];
    }
  }
}

// =====================================================================
extern "C" void kernel_launch(void* const* d_in, const int* in_sizes, int n_in,
                              void* d_out, int out_size, void* d_ws, size_t ws_size,
                              hipStream_t stream) {
  const float* x   = (const float*)d_in[0];
  float*       out = (float*)d_out;
  float*       ent = (float*)d_ws;   // 8192 floats = 32 KB scratch

  entropy_kernel<<<NBLOCKS, 256, 0, stream>>>(x, ent);
  upsample_kernel<<<NCH, 256, 0, stream>>>(ent, out);
}